// KGCN_27221502722624
// MI455X (gfx1250) — compile-verified
//
#include <hip/hip_runtime.h>
#include <hip/hip_bf16.h>

typedef __attribute__((ext_vector_type(2))) float v2f;
typedef __attribute__((ext_vector_type(8))) float v8f;

#define DD   128   // embedding dim
#define SS   16    // neighbors
#define STR  132   // padded LDS row stride (floats) -> conflict-free
#define TM   16    // batch rows per block

__global__ __launch_bounds__(256) void kgcn_fused(
    const int* __restrict__ u, const int* __restrict__ v,
    const int* __restrict__ adj_ent, const int* __restrict__ adj_rel,
    const float* __restrict__ usr_table, const float* __restrict__ ent_table,
    const float* __restrict__ rel_table, const float* __restrict__ W,
    const float* __restrict__ bvec, float* __restrict__ out, int B)
{
  __shared__ float s_u[TM * STR];    // u embeddings
  __shared__ float s_msg[TM * STR];  // attention message (reused each iter)
  __shared__ float s_x[TM * STR];    // GEMM input X = msg + rep  (then rep)

  const int tid  = threadIdx.x;
  const int lane = tid & 31;
  const int wave = tid >> 5;     // 0..7
  const int hi   = lane >> 4;    // 0/1 : K-pair select for WMMA frags
  const int lo   = lane & 15;    // M (A) / N (B,D) index
  const int row0 = blockIdx.x * TM;

  // ---------------- Phase 1: gather + attention + message (2 rows / wave) ---
  for (int rr = 0; rr < 2; ++rr) {
    int r   = wave * 2 + rr;
    int row = row0 + r; if (row >= B) row = B - 1;
    int uu = u[row], vv = v[row];

    // u embedding -> LDS, float4 per lane (coalesced 512B)
    float4 u4 = ((const float4*)(usr_table + (long)uu * DD))[lane];
    *(float4*)&s_u[r * STR + lane * 4] = u4;

    // neighbor indices (lanes 16-31 duplicate lanes 0-15)
    int eidx = adj_ent[(long)vv * SS + lo];
    int ridx = adj_rel[(long)vv * SS + lo];

    // scores (lanes 0..15): dot(u_emb, rel_table[ridx])
    float score = 0.f;
    if (lane < 16) {
      const float4* rp = (const float4*)(rel_table + (long)ridx * DD);
      const float4* uq = (const float4*)&s_u[r * STR];
      #pragma unroll 8
      for (int k = 0; k < DD / 4; ++k) {
        float4 a = uq[k], b4 = rp[k];
        score += a.x * b4.x + a.y * b4.y + a.z * b4.z + a.w * b4.w;
      }
    }
    // softmax across the 16-lane group (xor masks stay inside the group)
    float m = score;
    for (int mask = 1; mask < 16; mask <<= 1) m = fmaxf(m, __shfl_xor(m, mask));
    float e = __expf(score - m);
    float ssum = e;
    for (int mask = 1; mask < 16; mask <<= 1) ssum += __shfl_xor(ssum, mask);
    float attn = e / ssum;

    // message: all 32 lanes own 4 consecutive d's; 512B coalesced per neighbor
    float4 acc = make_float4(0.f, 0.f, 0.f, 0.f);
    #pragma unroll
    for (int s = 0; s < SS; ++s) {
      float a  = __shfl(attn, s);
      int   ei = __shfl(eidx, s);
      float4 nb = *(const float4*)(ent_table + (long)ei * DD + lane * 4);
      acc.x += a * nb.x; acc.y += a * nb.y; acc.z += a * nb.z; acc.w += a * nb.w;
    }
    float4 rp0 = *(const float4*)(ent_table + (long)vv * DD + lane * 4);
    *(float4*)&s_msg[r * STR + lane * 4] = acc;
    float4 x0 = make_float4(acc.x + rp0.x, acc.y + rp0.y,
                            acc.z + rp0.z, acc.w + rp0.w);
    *(float4*)&s_x[r * STR + lane * 4] = x0;
  }

  // ---- Preload this wave's B slab: W[:, 16*wave .. 16*wave+16) in VGPRs ----
  // B frag layout (4x16 f32): lane holds N=lo, v0 = W[kk][N], v1 = W[kk+1][N],
  // kk = 4k + 2*hi  (mirrors documented 16x4 A layout).
  const int col = wave * 16 + lo;
  float bw0[32], bw1[32];
  #pragma unroll
  for (int k = 0; k < 32; ++k) {
    int kk = 4 * k + 2 * hi;
    bw0[k] = W[kk * DD + col];
    bw1[k] = W[(kk + 1) * DD + col];
  }
  const float bias = bvec[col];

  __syncthreads();

  // ------- Phase 2: two fused (X @ W + b, ReLU) iterations via WMMA f32 -----
  for (int it = 0; it < 2; ++it) {
    v8f c = {};
    #pragma unroll
    for (int k = 0; k < 32; ++k) {
      int kk = 4 * k + 2 * hi;
      v2f A, Bf;
      A.x  = s_x[lo * STR + kk];       // A: lane=M, K-pair by half-wave
      A.y  = s_x[lo * STR + kk + 1];
      Bf.x = bw0[k];
      Bf.y = bw1[k];
      c = __builtin_amdgcn_wmma_f32_16x16x4_f32(false, A, false, Bf,
                                                (short)0, c, false, false);
    }
    __syncthreads();                   // all waves finished reading s_x
    #pragma unroll
    for (int j = 0; j < 8; ++j) {      // D layout: VGPR j -> M = j + 8*hi
      int rowm  = j + 8 * hi;
      float val = fmaxf(c[j] + bias, 0.f);
      // next X = msg + rep ; after last iter keep rep itself for the dot
      s_x[rowm * STR + col] = (it == 1) ? val : (s_msg[rowm * STR + col] + val);
    }
    __syncthreads();
  }

  // ---------------- Phase 3: sigmoid(dot(u_emb, rep)) ----------------------
  for (int rr = 0; rr < 2; ++rr) {
    int r = wave * 2 + rr;
    float p = 0.f;
    #pragma unroll
    for (int i = 0; i < 4; ++i)
      p += s_u[r * STR + lane * 4 + i] * s_x[r * STR + lane * 4 + i];
    for (int mask = 1; mask < 32; mask <<= 1) p += __shfl_xor(p, mask);
    int row = row0 + r;
    if (lane == 0 && row < B) out[row] = 1.f / (1.f + __expf(-p));
  }
}

extern "C" void kernel_launch(void* const* d_in, const int* in_sizes, int n_in,
                              void* d_out, int out_size, void* d_ws, size_t ws_size,
                              hipStream_t stream) {
  const int*   u         = (const int*)d_in[0];
  const int*   v         = (const int*)d_in[1];
  const int*   adj_ent   = (const int*)d_in[2];
  const int*   adj_rel   = (const int*)d_in[3];
  const float* usr_table = (const float*)d_in[4];
  const float* ent_table = (const float*)d_in[5];
  const float* rel_table = (const float*)d_in[6];
  const float* W         = (const float*)d_in[7];
  const float* bvec      = (const float*)d_in[8];
  float*       out       = (float*)d_out;

  int B = in_sizes[0];
  int blocks = (B + TM - 1) / TM;
  kgcn_fused<<<blocks, 256, 0, stream>>>(u, v, adj_ent, adj_rel,
                                         usr_table, ent_table, rel_table,
                                         W, bvec, out, B);
}